// FastAsyncGNN_54855322304748
// MI455X (gfx1250) — compile-verified
//
#include <hip/hip_runtime.h>

// ---------------------------------------------------------------------------
// GCN forward for MI455X (gfx1250, wave32).
// GEMMs: V_WMMA_F32_16X16X4_F32, branch-free inner loop (B pre-packed into a
// pair-interleaved, column-padded layout so A and B are both single b64 loads
// per WMMA step). Edge scatter (the HBM-bound part) uses coalesced float4
// gathers + contiguous f32 atomics resolved in the 192MB L2.
// ---------------------------------------------------------------------------

typedef __attribute__((ext_vector_type(2))) float v2f;
typedef __attribute__((ext_vector_type(8))) float v8f;

#define FHID 96
#define FQ   (FHID / 4)   // float4 chunks per feature row

// ---- tiny utility kernels --------------------------------------------------

__global__ void k_fill_f32(float* __restrict__ p, float v, int n) {
  int i = blockIdx.x * blockDim.x + threadIdx.x;
  if (i < n) p[i] = v;
}

__global__ void k_zero_f4(float4* __restrict__ p, int n4) {
  int i = blockIdx.x * blockDim.x + threadIdx.x;
  if (i < n4) p[i] = make_float4(0.f, 0.f, 0.f, 0.f);
}

__global__ void k_degree(const int* __restrict__ dst, float* __restrict__ deg, int E) {
  int e = blockIdx.x * blockDim.x + threadIdx.x;
  if (e < E) atomicAdd(&deg[dst[e]], 1.0f);
}

__global__ void k_dinv(const float* __restrict__ deg, float* __restrict__ dinv, int n) {
  int i = blockIdx.x * blockDim.x + threadIdx.x;
  if (i < n) dinv[i] = rsqrtf(deg[i]);   // deg >= 1 always (self-loops)
}

__global__ void k_norm(const int* __restrict__ src, const int* __restrict__ dst,
                       const float* __restrict__ dinv, float* __restrict__ norm, int E) {
  int e = blockIdx.x * blockDim.x + threadIdx.x;
  if (e < E) norm[e] = dinv[src[e]] * dinv[dst[e]];
}

// ---- weight packing: pair-interleave K, zero-pad columns to Ncp ------------
// Bp[((k>>1)*Ncp + col)*2 + (k&1)] = (col < Nc) ? B[k*Nc + col] : 0
// => in the GEMM, a lane's (k, k+1) B operands are one aligned 8-byte load.

__global__ void k_packB(const float* __restrict__ B, float* __restrict__ Bp,
                        int K, int Nc, int Ncp) {
  int i = blockIdx.x * blockDim.x + threadIdx.x;
  if (i >= K * Ncp) return;
  const int k   = i / Ncp;
  const int col = i - k * Ncp;
  const float v = (col < Nc) ? B[(size_t)k * Nc + col] : 0.f;
  Bp[((size_t)(k >> 1) * Ncp + col) * 2 + (k & 1)] = v;
}

// ---- fp32 WMMA GEMM: C[M,Nc] = A[M,K] @ B[K,*] (+ bias) --------------------
// One wave per 16x16 output tile. Branch-free inner loop:
//   b64 A load + b64 packed-B load + v_wmma_f32_16x16x4_f32.
// Wave-uniform tile guard keeps EXEC all-ones through every WMMA.

__global__ void k_gemm_wmma_f32(const float* __restrict__ A,
                                const float* __restrict__ Bp,   // packed
                                const float* __restrict__ bias, // may be null
                                float* __restrict__ C,
                                int M, int K, int Nc, int Ncp,
                                int Mtiles, int Ntiles) {
  const int wave = threadIdx.x >> 5;
  const int lane = threadIdx.x & 31;
  const int tile = blockIdx.x * (blockDim.x >> 5) + wave;
  if (tile >= Mtiles * Ntiles) return;        // wave-uniform exit

  const int mt = tile / Ntiles;
  const int nt = tile % Ntiles;
  const int m0 = mt * 16;
  const int n0 = nt * 16;

  const int l15  = lane & 15;
  const int half = lane >> 4;                 // 0: K=0,1  1: K=2,3
  const int koff = half << 1;
  const int col  = n0 + l15;                  // col < Ncp always (padded)

  const float* __restrict__ Ar = A + (size_t)(m0 + l15) * K + koff;
  // packed B: pair index kk = k/2 + half, element (kk*Ncp + col)*2
  const float* __restrict__ Br = Bp + ((size_t)half * Ncp + col) * 2;

  v8f acc = {};
  for (int k = 0; k < K; k += 4) {
    const v2f a = *(const v2f*)(Ar + k);
    const v2f b = *(const v2f*)(Br + (size_t)(k >> 1) * Ncp * 2);
    // (neg_a, A, neg_b, B, c_mod, C, reuse_a, reuse_b)
    acc = __builtin_amdgcn_wmma_f32_16x16x4_f32(false, a, false, b,
                                                (short)0, acc, false, false);
  }

  if (col < Nc) {                             // only the 40-wide head masks
    const float bv = bias ? bias[col] : 0.f;
    const int mbase = m0 + (half << 3);       // lanes 16-31 -> rows +8
    #pragma unroll
    for (int r = 0; r < 8; ++r)
      C[(size_t)(mbase + r) * Nc + col] = acc[r] + bv;
  }
}

// ---- edge scatter: agg[dst] += h[src] * norm[e] ----------------------------
// One thread per (edge, float4-chunk): contiguous 384B row reads and
// contiguous atomic bursts; hot rows live in the 192MB L2.

__global__ void k_scatter(const float4* __restrict__ h4,
                          const int* __restrict__ src,
                          const int* __restrict__ dst,
                          const float* __restrict__ norm,
                          float* __restrict__ agg, int E) {
  int i = blockIdx.x * blockDim.x + threadIdx.x;
  if (i >= E * FQ) return;
  const int e = i / FQ;
  const int q = i - e * FQ;
  const float  s = norm[e];
  const float4 v = h4[(size_t)src[e] * FQ + q];
  float* __restrict__ out = agg + (size_t)dst[e] * FHID + q * 4;
  atomicAdd(out + 0, v.x * s);
  atomicAdd(out + 1, v.y * s);
  atomicAdd(out + 2, v.z * s);
  atomicAdd(out + 3, v.w * s);
}

// ---- fused epilogue: h = relu(agg + tmp * dinv^2 + b) ----------------------
// Self-loop contribution folded in here so it never touches atomics.

__global__ void k_epilogue(const float4* __restrict__ agg4,
                           const float4* __restrict__ tmp4,
                           const float* __restrict__ dinv,
                           const float* __restrict__ bias,
                           float4* __restrict__ out4, int Nn) {
  int i = blockIdx.x * blockDim.x + threadIdx.x;
  if (i >= Nn * FQ) return;
  const int node = i / FQ;
  const int q    = i - node * FQ;
  const float d  = dinv[node];
  const float w  = d * d;
  const float4 a = agg4[i];
  const float4 t = tmp4[i];
  const int f = q * 4;
  float4 r;
  r.x = fmaxf(a.x + t.x * w + bias[f + 0], 0.f);
  r.y = fmaxf(a.y + t.y * w + bias[f + 1], 0.f);
  r.z = fmaxf(a.z + t.z * w + bias[f + 2], 0.f);
  r.w = fmaxf(a.w + t.w * w + bias[f + 3], 0.f);
  out4[i] = r;
}

// ---------------------------------------------------------------------------

static inline int cdiv(int a, int b) { return (a + b - 1) / b; }

extern "C" void kernel_launch(void* const* d_in, const int* in_sizes, int n_in,
                              void* d_out, int out_size, void* d_ws, size_t ws_size,
                              hipStream_t stream) {
  const float* x    = (const float*)d_in[0];
  const int*   eidx = (const int*)  d_in[1];
  const float* W1   = (const float*)d_in[2];
  const float* b1   = (const float*)d_in[3];
  const float* W2   = (const float*)d_in[4];
  const float* b2   = (const float*)d_in[5];
  const float* Wout = (const float*)d_in[6];
  const float* bout = (const float*)d_in[7];

  const int F_IN  = 128;
  const int F_OUT = 40;
  const int F_OUTP = 48;                    // padded to 16-col multiple
  const int N = in_sizes[0] / F_IN;
  const int E = in_sizes[1] / 2;
  const int* src = eidx;
  const int* dst = eidx + E;

  // workspace layout (floats); all offsets 16B-aligned
  float* deg   = (float*)d_ws;
  float* dinv  = deg   + N;
  float* norm  = dinv  + N;
  float* tmp   = norm  + E;                  // pre-aggregation GEMM out [N,96]
  float* agg   = tmp   + (size_t)N * FHID;   // scatter accumulator     [N,96]
  float* hbuf  = agg   + (size_t)N * FHID;   // layer activation        [N,96]
  float* W1p   = hbuf  + (size_t)N * FHID;   // packed weights
  float* W2p   = W1p   + (size_t)F_IN * FHID;
  float* Woutp = W2p   + (size_t)FHID * FHID;
  float* outp  = (float*)d_out;

  const int T = 256;
  const int NF4 = (N * FHID) / 4;

  // ---- degree / norm precompute + weight packing ----
  k_fill_f32<<<cdiv(N, T), T, 0, stream>>>(deg, 1.0f, N);          // self-loop
  k_degree  <<<cdiv(E, T), T, 0, stream>>>(dst, deg, E);
  k_dinv    <<<cdiv(N, T), T, 0, stream>>>(deg, dinv, N);
  k_norm    <<<cdiv(E, T), T, 0, stream>>>(src, dst, dinv, norm, E);
  k_packB<<<cdiv(F_IN * FHID,   T), T, 0, stream>>>(W1,   W1p,   F_IN, FHID,  FHID);
  k_packB<<<cdiv(FHID * FHID,   T), T, 0, stream>>>(W2,   W2p,   FHID, FHID,  FHID);
  k_packB<<<cdiv(FHID * F_OUTP, T), T, 0, stream>>>(Wout, Woutp, FHID, F_OUT, F_OUTP);

  const int Mtiles = cdiv(N, 16);
  const int wavesPerBlk = T / 32;

  // ---- layer 1: tmp = x@W1 ; agg = scatter ; hbuf = relu(agg+self+b1) ----
  {
    const int Ntl = FHID / 16;
    k_zero_f4<<<cdiv(NF4, T), T, 0, stream>>>((float4*)agg, NF4);
    k_gemm_wmma_f32<<<cdiv(Mtiles * Ntl, wavesPerBlk), T, 0, stream>>>(
        x, W1p, nullptr, tmp, N, F_IN, FHID, FHID, Mtiles, Ntl);
    k_scatter<<<cdiv(E * FQ, T), T, 0, stream>>>(
        (const float4*)tmp, src, dst, norm, agg, E);
    k_epilogue<<<cdiv(N * FQ, T), T, 0, stream>>>(
        (const float4*)agg, (const float4*)tmp, dinv, b1, (float4*)hbuf, N);
  }

  // ---- layer 2: tmp = hbuf@W2 ; agg ; hbuf = relu(...) (safe reuse) ----
  {
    const int Ntl = FHID / 16;
    k_zero_f4<<<cdiv(NF4, T), T, 0, stream>>>((float4*)agg, NF4);
    k_gemm_wmma_f32<<<cdiv(Mtiles * Ntl, wavesPerBlk), T, 0, stream>>>(
        hbuf, W2p, nullptr, tmp, N, FHID, FHID, FHID, Mtiles, Ntl);
    k_scatter<<<cdiv(E * FQ, T), T, 0, stream>>>(
        (const float4*)tmp, src, dst, norm, agg, E);
    k_epilogue<<<cdiv(N * FQ, T), T, 0, stream>>>(
        (const float4*)agg, (const float4*)tmp, dinv, b2, (float4*)hbuf, N);
  }

  // ---- head: out = hbuf@Wout + bout (padded B, store-guarded cols) ----
  {
    const int Ntl = F_OUTP / 16;
    k_gemm_wmma_f32<<<cdiv(Mtiles * Ntl, wavesPerBlk), T, 0, stream>>>(
        hbuf, Woutp, bout, outp, N, FHID, F_OUT, F_OUTP, Mtiles, Ntl);
  }
}